// DAMAS_FISTANet_69320772157705
// MI455X (gfx1250) — compile-verified
//
#include <hip/hip_runtime.h>

#define B_ 16
#define NM 56
#define G_ 1681
#define LAYERS_ 10

typedef __attribute__((ext_vector_type(2))) float v2f;
typedef __attribute__((ext_vector_type(8))) float v8f;
typedef __attribute__((ext_vector_type(4))) unsigned int u32x4;
typedef __attribute__((ext_vector_type(8))) int i32x8;
typedef __attribute__((ext_vector_type(4))) int i32x4;

// ---------------------------------------------------------------------------
// Kernel 1: DAS[b,g] = Re( x^H CSM_b x ) / NM^2,  x = wk_weight * wk[b,:,g]
// One thread per g; CSM and the per-thread x vectors staged in LDS.
// ---------------------------------------------------------------------------
__global__ __launch_bounds__(64) void das_kernel(
    const float* __restrict__ CSM_re, const float* __restrict__ CSM_im,
    const float* __restrict__ wk_re,  const float* __restrict__ wk_im,
    const float* __restrict__ wk_weight, float* __restrict__ DAS)
{
    __shared__ float sCr[NM * NM], sCi[NM * NM];
    __shared__ float sxr[NM * 64], sxi[NM * 64];
    const int b   = blockIdx.y;
    const int tid = threadIdx.x;
    const int g   = blockIdx.x * 64 + tid;
    const float ww = wk_weight[0];

    for (int i = tid; i < NM * NM; i += 64) {
        sCr[i] = CSM_re[b * NM * NM + i];
        sCi[i] = CSM_im[b * NM * NM + i];
    }
    const bool valid = (g < G_);
    const int  gg    = valid ? g : 0;
    for (int m = 0; m < NM; ++m) {
        float xr = wk_re[(size_t)b * NM * G_ + (size_t)m * G_ + gg] * ww;
        float xi = wk_im[(size_t)b * NM * G_ + (size_t)m * G_ + gg] * ww;
        sxr[m * 64 + tid] = valid ? xr : 0.f;
        sxi[m * 64 + tid] = valid ? xi : 0.f;
    }
    __syncthreads();

    float acc = 0.f;
    for (int m = 0; m < NM; ++m) {
        float vr = 0.f, vi = 0.f;
        for (int n = 0; n < NM; ++n) {
            float cr = sCr[m * NM + n], ci = sCi[m * NM + n];
            float xr = sxr[n * 64 + tid], xi = sxi[n * 64 + tid];
            vr = fmaf(cr, xr, vr); vr = fmaf(-ci, xi, vr);
            vi = fmaf(cr, xi, vi); vi = fmaf( ci, xr, vi);
        }
        float xr = sxr[m * 64 + tid], xi = sxi[m * 64 + tid];
        acc = fmaf(xr, vr, acc);   // Re(conj(x_m) * v_m)
        acc = fmaf(xi, vi, acc);
    }
    if (valid) DAS[b * G_ + g] = acc * (1.0f / (float)(NM * NM));
}

// ---------------------------------------------------------------------------
// Kernel 2: ATb[b,g] = sum_j A_K[b,j,g] * DAS[b,j]  (coalesced over g).
// A_K is a one-shot 180.8MB stream -> non-temporal loads so it does not
// evict ATA_K residency in the 192MB L2 used by the 10 FISTA iterations.
// Also zero-initializes the (x, y) ping-pong state.
// ---------------------------------------------------------------------------
__global__ __launch_bounds__(256) void atb_kernel(
    const float* __restrict__ A_K, const float* __restrict__ DAS,
    float* __restrict__ ATb, float* __restrict__ x0, float* __restrict__ y0)
{
    __shared__ float sd[256];
    const int b   = blockIdx.y;
    const int tid = threadIdx.x;
    const int g   = blockIdx.x * 256 + tid;
    const size_t base = (size_t)b * G_ * G_;
    float acc = 0.f;
    for (int j0 = 0; j0 < G_; j0 += 256) {
        int j = j0 + tid;
        sd[tid] = (j < G_) ? DAS[b * G_ + j] : 0.f;
        __syncthreads();
        int lim = min(256, G_ - j0);
        if (g < G_) {
            for (int jj = 0; jj < lim; ++jj) {
                float a = __builtin_nontemporal_load(&A_K[base + (size_t)(j0 + jj) * G_ + g]);
                acc = fmaf(a, sd[jj], acc);
            }
        }
        __syncthreads();
    }
    if (g < G_) {
        ATb[b * G_ + g] = acc;
        x0 [b * G_ + g] = 0.f;
        y0 [b * G_ + g] = 0.f;
    }
}

// ---------------------------------------------------------------------------
// Kernel 3: one FISTA layer.
//   r    = ys*y - invL*lam*(ATA_K @ y) + invL*lam*ATb
//   xnew = relu(r);  ynew = w_rho*xnew + b_rho*(xnew - xold)
// The whole y[b] vector (6724 B) is DMA'd into LDS by the Tensor Data Mover
// (one tensor_load_to_lds issued by wave 0, D# built per ISA §8.3/8.4:
// 1-row tile, data_size=4B, tensor_dim0=tile_dim0=1681, stride=1681).
// Then each thread streams one ATA row (L2-resident across iterations).
// ---------------------------------------------------------------------------
__global__ __launch_bounds__(128) void fista_kernel(
    const float* __restrict__ ATA, const float* __restrict__ ATb,
    const float* __restrict__ L_K,
    const float* __restrict__ lambda_step, const float* __restrict__ y_step,
    const float* __restrict__ w_rho, const float* __restrict__ b_rho,
    const float* __restrict__ y_in, const float* __restrict__ x_in,
    float* __restrict__ y_out, float* __restrict__ x_out, int layer)
{
    __shared__ float sy[1684];           // 1681 used, padded to 16B multiple
    const int b   = blockIdx.y;
    const int tid = threadIdx.x;
    const int g   = blockIdx.x * 128 + tid;
    const float* __restrict__ yrow = y_in + (size_t)b * G_;

#if __has_builtin(__builtin_amdgcn_tensor_load_to_lds) && __has_builtin(__builtin_amdgcn_s_wait_tensorcnt)
    if (tid < 32) {                      // wave 0 issues the TDM DMA
        unsigned long long ga = (unsigned long long)(uintptr_t)yrow;
        unsigned int lds_off  = (unsigned int)(uintptr_t)&sy[0]; // low32 of flat == LDS byte offset
        u32x4 d0;
        d0[0] = 1u;                                            // count=1, user desc
        d0[1] = lds_off;                                       // lds_addr
        d0[2] = (unsigned int)(ga & 0xFFFFFFFFull);            // global_addr[31:0]
        d0[3] = (unsigned int)((ga >> 32) & 0x1FFFFFFull)      // global_addr[56:32]
              | (2u << 30);                                    // type = 2 ("image")
        i32x8 d1;
        d1[0] = (2 << 16);            // data_size = 2 (4 bytes); wg_mask = 0
        d1[1] = (int)((unsigned)G_ << 16);  // tensor_dim0[15:0] @ bits 63:48
        d1[2] = (1 << 16);            // tensor_dim0[31:16]=0; tensor_dim1=1 @ bits 95:80
        d1[3] = (int)((unsigned)G_ << 16);  // tile_dim0 @ bits 127:112
        d1[4] = 1;                    // tile_dim1 = 1; tile_dim2 = 0
        d1[5] = G_;                   // tensor_dim0_stride[31:0]
        d1[6] = 0;                    // stride0[47:32]=0; stride1[15:0]=0
        d1[7] = 0;                    // stride1[47:16]=0
        i32x4 d2 = {0, 0, 0, 0};
        i32x4 d3 = {0, 0, 0, 0};
        i32x8 d4 = {0, 0, 0, 0, 0, 0, 0, 0};
        __builtin_amdgcn_tensor_load_to_lds(d0, d1, d2, d3, d4, 0);
        __builtin_amdgcn_s_wait_tensorcnt((short)0);
    }
    __syncthreads();
#else
    for (int j = tid; j < G_; j += 128) sy[j] = yrow[j];
    __syncthreads();
#endif

    const float* __restrict__ row =
        ATA + (size_t)b * G_ * G_ + (size_t)((g < G_) ? g : 0) * G_;
    float acc = 0.f;
    int j0 = 0;
    for (; j0 + 256 <= G_; j0 += 256) {
        if (j0 + 512 <= G_) __builtin_prefetch(row + j0 + 256, 0, 0);
        #pragma unroll 8
        for (int jj = 0; jj < 256; ++jj)
            acc = fmaf(row[j0 + jj], sy[j0 + jj], acc);
    }
    for (int j = j0; j < G_; ++j)
        acc = fmaf(row[j], sy[j], acc);

    if (g < G_) {
        float invL = 1.0f / L_K[b];
        float lam  = lambda_step[layer];
        float ys   = y_step[layer];
        float yv   = yrow[g];
        float r    = ys * yv - invL * lam * acc + invL * lam * ATb[b * G_ + g];
        float xn   = fmaxf(r, 0.f);
        float yn   = w_rho[0] * xn + b_rho[0] * (xn - x_in[b * G_ + g]);
        x_out[b * G_ + g] = xn;
        y_out[b * G_ + g] = yn;
    }
}

// ---------------------------------------------------------------------------
// Kernel 4: final linear layer via V_WMMA_F32_16X16X4_F32 (fp32-exact).
//   z[b,n]   = sum_k X[b,k] * W[n,k] + bias[n]        (M=16 == WMMA tile M)
//   out[b,n] = relu(w1*z + w2*X[b,n])
// Guard-free main loop (K=0..1679): column n of D depends only on column n
// of B, so clamped B rows for the N-tail are harmless (never stored), and
// the K-tail is handled in ONE masked step where only A is zeroed.
// ---------------------------------------------------------------------------
__global__ __launch_bounds__(128) void linear_wmma_kernel(
    const float* __restrict__ X, const float* __restrict__ W,
    const float* __restrict__ bias,
    const float* __restrict__ w1, const float* __restrict__ w2,
    float* __restrict__ out)
{
    const int lane = threadIdx.x & 31;
    const int wave = threadIdx.x >> 5;
    const int tile = blockIdx.x * 4 + wave;
    const int n0   = tile * 16;
    if (n0 >= G_) return;                 // wave-uniform exit

    const int  half    = lane >> 4;       // 0 or 1
    const int  sub     = lane & 15;       // M index (A) / N index (B)
    const int  n       = n0 + sub;
    const bool ncol_ok = (n < G_);
    const int  nc      = ncol_ok ? n : 0;

    const float* __restrict__ rowX = X + sub * G_;            // A: X[m=sub, :]
    const float* __restrict__ rowW = W + (size_t)nc * G_;     // B: W[n, :]

    v8f c = {};
    const int KMAIN = (G_ / 4) * 4;       // 1680: unguarded main loop
    for (int k0 = 0; k0 < KMAIN; k0 += 4) {
        const int ka = k0 + half * 2;
        v2f a, bf;
        a[0]  = rowX[ka];
        a[1]  = rowX[ka + 1];
        bf[0] = rowW[ka];
        bf[1] = rowW[ka + 1];
        c = __builtin_amdgcn_wmma_f32_16x16x4_f32(
                false, a, false, bf, (short)0, c, false, false);
    }
    {   // K tail: k in [1680,1684), only k=1680 valid -> zero A only.
        const int  ka  = KMAIN + half * 2;
        const bool ok0 = (ka < G_);
        const bool ok1 = (ka + 1 < G_);
        v2f a, bf;
        float a0 = rowX[ok0 ? ka : 0];
        float a1 = rowX[ok1 ? (ka + 1) : 0];
        a[0]  = ok0 ? a0 : 0.f;
        a[1]  = ok1 ? a1 : 0.f;
        bf[0] = rowW[ok0 ? ka : 0];       // finite garbage OK: a==0 there
        bf[1] = rowW[ok1 ? (ka + 1) : 0];
        c = __builtin_amdgcn_wmma_f32_16x16x4_f32(
                false, a, false, bf, (short)0, c, false, false);
    }

    if (ncol_ok) {
        const float s1 = w1[0], s2 = w2[0];
        const float bb = bias[n];
        #pragma unroll
        for (int i = 0; i < 8; ++i) {
            int   m = half * 8 + i;       // batch index 0..15
            float z = c[i] + bb;
            float t = X[m * G_ + n];      // temp = xnew
            float o = s1 * z + s2 * t;
            out[m * G_ + n] = fmaxf(o, 0.f);
        }
    }
}

// ---------------------------------------------------------------------------
// Host launcher (graph-capture safe: only launches on `stream`).
// Inputs: 0 CSM_re, 1 CSM_im, 2 wk_re, 3 wk_im, 4 A_K, 5 ATA_K, 6 L_K,
//         7 lambda_step, 8 y_step, 9 wk_weight, 10 w_rho, 11 b_rho,
//         12 linear_W, 13 linear_b, 14 weight1, 15 weight2
// ---------------------------------------------------------------------------
extern "C" void kernel_launch(void* const* d_in, const int* in_sizes, int n_in,
                              void* d_out, int out_size, void* d_ws, size_t ws_size,
                              hipStream_t stream)
{
    const float* CSM_re      = (const float*)d_in[0];
    const float* CSM_im      = (const float*)d_in[1];
    const float* wk_re       = (const float*)d_in[2];
    const float* wk_im       = (const float*)d_in[3];
    const float* A_K         = (const float*)d_in[4];
    const float* ATA_K       = (const float*)d_in[5];
    const float* L_K         = (const float*)d_in[6];
    const float* lambda_step = (const float*)d_in[7];
    const float* y_step      = (const float*)d_in[8];
    const float* wk_weight   = (const float*)d_in[9];
    const float* w_rho       = (const float*)d_in[10];
    const float* b_rho       = (const float*)d_in[11];
    const float* linear_W    = (const float*)d_in[12];
    const float* linear_b    = (const float*)d_in[13];
    const float* weight1     = (const float*)d_in[14];
    const float* weight2     = (const float*)d_in[15];

    float* ws  = (float*)d_ws;
    const size_t BG = (size_t)B_ * G_;
    float* DAS = ws + 0 * BG;
    float* ATb = ws + 1 * BG;
    float* xa  = ws + 2 * BG;
    float* ya  = ws + 3 * BG;
    float* xb  = ws + 4 * BG;
    float* yb  = ws + 5 * BG;

    dim3 g1((G_ + 63) / 64, B_);
    das_kernel<<<g1, 64, 0, stream>>>(CSM_re, CSM_im, wk_re, wk_im, wk_weight, DAS);

    dim3 g2((G_ + 255) / 256, B_);
    atb_kernel<<<g2, 256, 0, stream>>>(A_K, DAS, ATb, xa, ya);

    dim3 g3((G_ + 127) / 128, B_);
    for (int l = 0; l < LAYERS_; ++l) {
        const float* yi = (l & 1) ? yb : ya;
        const float* xi = (l & 1) ? xb : xa;
        float*       yo = (l & 1) ? ya : yb;
        float*       xo = (l & 1) ? xa : xb;
        fista_kernel<<<g3, 128, 0, stream>>>(ATA_K, ATb, L_K, lambda_step, y_step,
                                             w_rho, b_rho, yi, xi, yo, xo, l);
    }
    // LAYERS_=10: last layer (l=9, odd) wrote x into xa.
    const float* xfinal = xa;

    const int ntiles = (G_ + 15) / 16;           // 106
    dim3 g4((ntiles + 3) / 4);                   // 4 waves per block
    linear_wmma_kernel<<<g4, 128, 0, stream>>>(xfinal, linear_W, linear_b,
                                               weight1, weight2, (float*)d_out);
}